// BilinearSampler_10479720202258
// MI455X (gfx1250) — compile-verified
//
#include <hip/hip_runtime.h>

// Bilinear grid sampler, B=16 H=256 W=256 C=64, align_corners=True semantics.
// Memory-bound gather: optimized for B128 vector memory ops + NT cache hints.

typedef float v4f __attribute__((ext_vector_type(4)));
typedef float v2f __attribute__((ext_vector_type(2)));

constexpr int Bn = 16, Hn = 256, Wn = 256, Cn = 64;
constexpr int C4 = Cn / 4;            // 16 float4 chunks per pixel
constexpr int PIX = Bn * Hn * Wn;     // 1,048,576 pixels

__global__ __launch_bounds__(256) void bilinear_sampler_kernel(
    const float* __restrict__ U,
    const float* __restrict__ grid,
    float* __restrict__ out)
{
    const int tid    = blockIdx.x * 256 + (int)threadIdx.x;
    const int lane_c = tid & (C4 - 1);   // which float4 chunk of the 64 channels
    const int p      = tid >> 4;         // pixel index in [0, B*H*W)

    // grid coords for this pixel (read-once -> non-temporal)
    const v2f g = __builtin_nontemporal_load((const v2f*)grid + p);
    const float x = 0.5f * ((g.x + 1.0f) * (float)(Wn - 1));
    const float y = 0.5f * ((g.y + 1.0f) * (float)(Hn - 1));

    const float xfl = floorf(x);
    const float yfl = floorf(y);
    const int x0 = (int)xfl;
    const int y0 = (int)yfl;
    const int x1 = x0 + 1;
    const int y1 = y0 + 1;

    const int x0c = min(max(x0, 0), Wn - 1);
    const int x1c = min(max(x1, 0), Wn - 1);
    const int y0c = min(max(y0, 0), Hn - 1);
    const int y1c = min(max(y1, 0), Hn - 1);

    const float x0f = (float)x0c, x1f = (float)x1c;
    const float y0f = (float)y0c, y1f = (float)y1c;

    const float wa = (x1f - x) * (y1f - y);
    const float wb = (x1f - x) * (y - y0f);
    const float wc = (x - x0f) * (y1f - y);
    const float wd = (x - x0f) * (y - y0f);

    const int b    = p / (Hn * Wn);
    const int base = b * (Hn * Wn);

    // addresses in float4 units; each pixel row of U is C4=16 float4s
    const v4f* __restrict__ Uv = (const v4f*)U;
    const int ia = (base + y0c * Wn + x0c) * C4 + lane_c;
    const int ib = (base + y1c * Wn + x0c) * C4 + lane_c;
    const int ic = (base + y0c * Wn + x1c) * C4 + lane_c;
    const int id = (base + y1c * Wn + x1c) * C4 + lane_c;

    // four B128 gathers (U stays RT-cached: heavy L2 reuse, 16MB/image footprint)
    const v4f Ia = Uv[ia];
    const v4f Ib = Uv[ib];
    const v4f Ic = Uv[ic];
    const v4f Id = Uv[id];

    const v4f r = wa * Ia + wb * Ib + wc * Ic + wd * Id;

    // streaming output: non-temporal B128 store, keep L2 for U
    __builtin_nontemporal_store(r, (v4f*)out + (p * C4 + lane_c));
}

extern "C" void kernel_launch(void* const* d_in, const int* in_sizes, int n_in,
                              void* d_out, int out_size, void* d_ws, size_t ws_size,
                              hipStream_t stream)
{
    const float* U    = (const float*)d_in[0];
    const float* grid = (const float*)d_in[1];
    float*       out  = (float*)d_out;

    const int total_threads = PIX * C4;          // 16 threads per pixel
    const int block = 256;                       // 8 wave32s
    const int nblocks = total_threads / block;   // 65,536

    bilinear_sampler_kernel<<<dim3(nblocks), dim3(block), 0, stream>>>(U, grid, out);
}